// HHGNN_36481452212904
// MI455X (gfx1250) — compile-verified
//
#include <hip/hip_runtime.h>
#include <hip/hip_bf16.h>

// ---------------- problem constants (from reference) ----------------
#define N_PER_TYPE 500000
#define N_NODES   1000000
#define N_EDGES   2000000
#define NNZ       6000000   // 3 * N_EDGES
#define IN_DIM    16
#define FEAT      3
#define CLS       13
#define OUTD      2

typedef __attribute__((ext_vector_type(2))) float v2f;
typedef __attribute__((ext_vector_type(8))) float v8f;

static inline int cdiv(long a, int b) { return (int)((a + b - 1) / b); }
static __device__ __forceinline__ int imin(int a, int b) { return a < b ? a : b; }

// ---------------- utility: zero a float region ----------------
__global__ void k_zero(float* __restrict__ p, int n) {
    int i = blockIdx.x * blockDim.x + threadIdx.x;
    if (i < n) p[i] = 0.0f;
}

// ---------------- per-type input projection: x = [xa@Wa+ba ; xb@Wb+bb] ----------------
__global__ __launch_bounds__(256)
void k_project(const float* __restrict__ xa, const float* __restrict__ xb,
               const float* __restrict__ Wa, const float* __restrict__ ba,
               const float* __restrict__ Wb, const float* __restrict__ bb,
               float* __restrict__ xbuf) {
    int n = blockIdx.x * blockDim.x + threadIdx.x;
    if (n >= N_NODES) return;
    const float* src;
    const float* W;
    const float* b;
    if (n < N_PER_TYPE) { src = xa + (long)n * IN_DIM; W = Wa; b = ba; }
    else                { src = xb + (long)(n - N_PER_TYPE) * IN_DIM; W = Wb; b = bb; }
    float o0 = b[0], o1 = b[1], o2 = b[2];
#pragma unroll
    for (int i = 0; i < IN_DIM; ++i) {
        float v = src[i];
        o0 += v * W[i * FEAT + 0];
        o1 += v * W[i * FEAT + 1];
        o2 += v * W[i * FEAT + 2];
    }
    xbuf[n * 3 + 0] = o0;
    xbuf[n * 3 + 1] = o1;
    xbuf[n * 3 + 2] = o2;
}

// ---------------- node degree via float atomics ----------------
__global__ __launch_bounds__(256)
void k_degree(const int* __restrict__ node_idx, float* __restrict__ dn) {
    int i = blockIdx.x * blockDim.x + threadIdx.x;
    if (i >= NNZ) return;
    unsafeAtomicAdd(&dn[node_idx[i]], 1.0f);
}

// ---------------- xt = x @ theta (3x3), in place ----------------
__global__ __launch_bounds__(256)
void k_theta(float* __restrict__ xbuf, const float* __restrict__ theta) {
    int n = blockIdx.x * blockDim.x + threadIdx.x;
    if (n >= N_NODES) return;
    float r0 = xbuf[n * 3 + 0], r1 = xbuf[n * 3 + 1], r2 = xbuf[n * 3 + 2];
#pragma unroll
    for (int c = 0; c < 3; ++c) {
        xbuf[n * 3 + c] = r0 * theta[0 * 3 + c] + r1 * theta[1 * 3 + c] + r2 * theta[2 * 3 + c];
    }
}

// ---------------- per-edge: e_feat = (xt[n0]+xt[n1]+xt[n2])/3 ; scatter to acc ----------------
__global__ __launch_bounds__(256)
void k_edge_scatter(const int* __restrict__ node_idx, const float* __restrict__ xt,
                    float* __restrict__ acc) {
    int e = blockIdx.x * blockDim.x + threadIdx.x;
    if (e >= N_EDGES) return;
    int base = 3 * e;
    int n0 = node_idx[base + 0];
    int n1 = node_idx[base + 1];
    int n2 = node_idx[base + 2];
    float s[3];
#pragma unroll
    for (int c = 0; c < 3; ++c)
        s[c] = (xt[n0 * 3 + c] + xt[n1 * 3 + c] + xt[n2 * 3 + c]) * (1.0f / 3.0f);
#pragma unroll
    for (int c = 0; c < 3; ++c) {
        unsafeAtomicAdd(&acc[n0 * 3 + c], s[c]);
        unsafeAtomicAdd(&acc[n1 * 3 + c], s[c]);
        unsafeAtomicAdd(&acc[n2 * 3 + c], s[c]);
    }
}

// ---------------- x_next = relu(acc * D^-1 + bias) ----------------
__global__ __launch_bounds__(256)
void k_finalize(const float* __restrict__ acc, const float* __restrict__ dn,
                const float* __restrict__ bias, float* __restrict__ xbuf) {
    int n = blockIdx.x * blockDim.x + threadIdx.x;
    if (n >= N_NODES) return;
    float d = dn[n];
    float inv = (d > 0.0f) ? (1.0f / d) : 0.0f;
#pragma unroll
    for (int c = 0; c < 3; ++c) {
        float v = acc[n * 3 + c] * inv + bias[c];
        xbuf[n * 3 + c] = fmaxf(v, 0.0f);
    }
}

// ---------------- edge MLP with V_WMMA_F32_16X16X4_F32 ----------------
// One wave = one 16-edge tile.  Layer 1: ef[16x9] (K pad 12) @ W1pad[16x16],
// 3 K=4 WMMAs.  Layer 2: h1 @ W2pad[16x16], 4 K=4 WMMAs, A re-layouted through
// an LDS tile.  Layer 3 (13->2) scalar from LDS.  Weights are staged once per
// block into LDS, zero-padded and pair-packed so each B fragment is a single
// aligned ds_load_b64 and no exec-masked global loads remain in the hot path.
__global__ __launch_bounds__(256)
void k_edge_mlp(const int* __restrict__ nidx, const float* __restrict__ x,
                const float* __restrict__ W1, const float* __restrict__ b1,
                const float* __restrict__ W2, const float* __restrict__ b2,
                const float* __restrict__ W3, const float* __restrict__ b3,
                float* __restrict__ out) {
    __shared__ float w1p[8][16][2];    // [k>>1][n][k&1], zero-padded 16x16 W1
    __shared__ float w2p[8][16][2];    // [k>>1][n][k&1], zero-padded 16x16 W2
    __shared__ float b1p[16], b2p[16]; // zero-padded biases
    __shared__ float w3p[16][2];       // zero-padded W3
    __shared__ float b3p[2];
    __shared__ float tile[8][16][18];  // per-wave 16x16 h tile (stride 18: b64-aligned)

    const int tid = threadIdx.x;

    // ---- stage zero-padded weights (clamped addresses: no exec-masked loads) ----
    {
        int k = tid >> 4, n = tid & 15;
        int kc1 = imin(k, 8), kc2 = imin(k, CLS - 1), ncl = imin(n, CLS - 1);
        float v1 = W1[kc1 * CLS + ncl];
        float v2 = W2[kc2 * CLS + ncl];
        w1p[k >> 1][n][k & 1] = (k < 9 && n < CLS) ? v1 : 0.0f;
        w2p[k >> 1][n][k & 1] = (k < CLS && n < CLS) ? v2 : 0.0f;
        if (tid < 16) {
            int t = imin(tid, CLS - 1);
            b1p[tid] = (tid < CLS) ? b1[t] : 0.0f;
            b2p[tid] = (tid < CLS) ? b2[t] : 0.0f;
        }
        if (tid < 32) {
            int r = tid >> 1, o = tid & 1;
            w3p[r][o] = (r < CLS) ? W3[imin(r, CLS - 1) * OUTD + o] : 0.0f;
        }
        if (tid < 2) b3p[tid] = b3[tid];
    }
    __syncthreads();

    const int  lane = tid & 31;
    const int  wave = tid >> 5;
    const int  m    = lane & 15;       // A-frag row / B-frag column
    const bool hi   = lane >= 16;      // K-pair selector: {0,1} vs {2,3}
    const int  edge = blockIdx.x * 128 + wave * 16 + m;

    const int base = 3 * edge;
    const int n0 = nidx[base + 0];
    const int n1 = nidx[base + 1];
    const int n2 = nidx[base + 2];

    // ---- A fragments for layer 1: ef(k) = x[node(k/3)*3 + k%3], k<9 else 0.
    // Per chunk c the lane's K pair is (4c+2*hi, 4c+2*hi+1); resolve the
    // (k -> node,comp) map by hand so each element is ONE address-selected load.
    v2f a0, a1, a2;
    a0.x = x[n0 * 3 + (hi ? 2 : 0)];                 // k=0 | k=2
    { int nn = hi ? n1 : n0; a0.y = x[nn * 3 + (hi ? 0 : 1)]; }  // k=3 | k=1
    { int nn = hi ? n2 : n1; a1.x = x[nn * 3 + (hi ? 0 : 1)]; }  // k=6 | k=4
    { int nn = hi ? n2 : n1; a1.y = x[nn * 3 + (hi ? 1 : 2)]; }  // k=7 | k=5
    {
        float t8 = x[n2 * 3 + 2];                    // k=8
        a2.x = hi ? 0.0f : t8;                       // k=10 is pad
        a2.y = 0.0f;                                 // k=9 | k=11: pad
    }

    // ---- layer 1: h1 = relu(ef @ W1 + b1) ----
    v8f acc = {};
    {
        v2f b = *(const v2f*)&w1p[0 + (hi ? 1 : 0)][m][0];
        acc = __builtin_amdgcn_wmma_f32_16x16x4_f32(false, a0, false, b, (short)0, acc, false, false);
        b = *(const v2f*)&w1p[2 + (hi ? 1 : 0)][m][0];
        acc = __builtin_amdgcn_wmma_f32_16x16x4_f32(false, a1, false, b, (short)0, acc, false, false);
        b = *(const v2f*)&w1p[4 + (hi ? 1 : 0)][m][0];
        acc = __builtin_amdgcn_wmma_f32_16x16x4_f32(false, a2, false, b, (short)0, acc, false, false);
    }
    {
        // D vgpr r: row M = r + 8*hi, col N = m.  Cols >= 13 are exactly 0
        // (zero B columns + zero bias), so no masking needed.
        float bias = b1p[m];
#pragma unroll
        for (int r = 0; r < 8; ++r)
            tile[wave][r + (hi ? 8 : 0)][m] = fmaxf(acc[r] + bias, 0.0f);
    }

    // ---- layer 2: h2 = relu(h1 @ W2 + b2), K = 13 padded to 16 ----
    v8f acc2 = {};
#pragma unroll
    for (int c = 0; c < 4; ++c) {
        int k0 = 4 * c + (hi ? 2 : 0);                       // even -> b64-aligned
        v2f a = *(const v2f*)&tile[wave][m][k0];
        v2f b = *(const v2f*)&w2p[2 * c + (hi ? 1 : 0)][m][0];
        acc2 = __builtin_amdgcn_wmma_f32_16x16x4_f32(false, a, false, b, (short)0, acc2, false, false);
    }
    {
        float bias = b2p[m];
#pragma unroll
        for (int r = 0; r < 8; ++r)
            tile[wave][r + (hi ? 8 : 0)][m] = fmaxf(acc2[r] + bias, 0.0f);
    }

    // ---- layer 3: out = h2 @ W3 + b3 (16 edges x 2 outputs = 1 per lane) ----
    {
        int me = lane >> 1;      // edge row 0..15
        int o  = lane & 1;       // output column
        float s = b3p[o];
#pragma unroll
        for (int k = 0; k < CLS; ++k) s += tile[wave][me][k] * w3p[k][o];
        int eo = blockIdx.x * 128 + wave * 16 + me;
        out[(long)eo * OUTD + o] = s;
    }
}

// ---------------- host-side launch ----------------
extern "C" void kernel_launch(void* const* d_in, const int* in_sizes, int n_in,
                              void* d_out, int out_size, void* d_ws, size_t ws_size,
                              hipStream_t stream) {
    (void)in_sizes; (void)n_in; (void)out_size; (void)ws_size;
    const float* x_a    = (const float*)d_in[0];
    const float* x_b    = (const float*)d_in[1];
    const int*   hyper  = (const int*)d_in[2];
    const int*   nodeix = hyper;                 // row 0 of hyperedge_index
    const float* W_a    = (const float*)d_in[3];
    const float* b_a    = (const float*)d_in[4];
    const float* W_b    = (const float*)d_in[5];
    const float* b_b    = (const float*)d_in[6];
    const float* theta1 = (const float*)d_in[7];
    const float* bias1  = (const float*)d_in[8];
    const float* theta2 = (const float*)d_in[9];
    const float* bias2  = (const float*)d_in[10];
    const float* W1     = (const float*)d_in[11];
    const float* b1     = (const float*)d_in[12];
    const float* W2     = (const float*)d_in[13];
    const float* b2     = (const float*)d_in[14];
    const float* W3     = (const float*)d_in[15];
    const float* b3     = (const float*)d_in[16];
    float* outp = (float*)d_out;

    // workspace layout (floats): xbuf[3M] | acc[3M] | dn[1M]  -> 28 MB
    float* xbuf = (float*)d_ws;
    float* acc  = xbuf + (long)3 * N_NODES;
    float* dn   = acc  + (long)3 * N_NODES;

    const int B = 256;

    // zero acc + dn (contiguous 4M floats)
    k_zero<<<cdiv((long)4 * N_NODES, B), B, 0, stream>>>(acc, 4 * N_NODES);
    // input projection
    k_project<<<cdiv(N_NODES, B), B, 0, stream>>>(x_a, x_b, W_a, b_a, W_b, b_b, xbuf);
    // node degrees
    k_degree<<<cdiv(NNZ, B), B, 0, stream>>>(nodeix, dn);

    // conv layer 1
    k_theta<<<cdiv(N_NODES, B), B, 0, stream>>>(xbuf, theta1);
    k_edge_scatter<<<cdiv(N_EDGES, B), B, 0, stream>>>(nodeix, xbuf, acc);
    k_finalize<<<cdiv(N_NODES, B), B, 0, stream>>>(acc, dn, bias1, xbuf);

    // conv layer 2
    k_zero<<<cdiv((long)3 * N_NODES, B), B, 0, stream>>>(acc, 3 * N_NODES);
    k_theta<<<cdiv(N_NODES, B), B, 0, stream>>>(xbuf, theta2);
    k_edge_scatter<<<cdiv(N_EDGES, B), B, 0, stream>>>(nodeix, xbuf, acc);
    k_finalize<<<cdiv(N_NODES, B), B, 0, stream>>>(acc, dn, bias2, xbuf);

    // edge MLP with WMMA: 128 edges per 256-thread block, exact division
    k_edge_mlp<<<N_EDGES / 128, B, 0, stream>>>(nodeix, xbuf, W1, b1, W2, b2, W3, b3, outp);
}